// ReferenceEncoderGST_55250459295776
// MI455X (gfx1250) — compile-verified
//
#include <hip/hip_runtime.h>
#include <hip/hip_bf16.h>

typedef __attribute__((ext_vector_type(2))) float v2f;
typedef __attribute__((ext_vector_type(8))) float v8f;

__device__ __forceinline__ v8f wmma4(v2f a, v2f b, v8f c) {
  // V_WMMA_F32_16X16X4_F32: D = A(16x4) * B(4x16) + C, fp32 throughout.
  return __builtin_amdgcn_wmma_f32_16x16x4_f32(false, a, false, b, (short)0, c,
                                               false, false);
}

__device__ __forceinline__ float sigmoidf_(float x) {
  return 1.0f / (1.0f + expf(-x));
}

#define V8ZERO {0.f,0.f,0.f,0.f,0.f,0.f,0.f,0.f}

// ---------------------------------------------------------------------------
// Fused Conv3x3 stride2 'SAME' + bias + BN(inference) + ReLU, implicit GEMM.
// One wave computes a 16(pixels) x 32(cout) fp32 tile = two WMMA C tiles that
// share a single A fetch (halves A traffic, 3 vmem per WMMA instead of 4).
// Layout NHWC. A-matrix lane layout (ISA 7.12.2): lane l (0..15) row=l K={0,1},
// lanes 16..31 row=l K={2,3}. B/D: row striped across lanes within a VGPR.
// ---------------------------------------------------------------------------
template <int CIN, int COUT>
__global__ __launch_bounds__(256) void conv_bn_relu_wmma(
    const float* __restrict__ in, float* __restrict__ out,
    const float* __restrict__ w,      // [3,3,CIN,COUT]
    const float* __restrict__ bias,   // [COUT]
    const float* __restrict__ gamma, const float* __restrict__ beta,
    const float* __restrict__ mean, const float* __restrict__ var,
    int Ti, int Mi, int To, int Mo, int padT, int padM, int pixTiles) {
  const int wave = (blockIdx.x * blockDim.x + threadIdx.x) >> 5;
  const int coutPairs = COUT >> 5;  // COUT >= 32 for all layers
  const int nTiles = pixTiles * coutPairs;
  if (wave >= nTiles) return;

  const int lane = threadIdx.x & 31;
  const int l = lane & 15;
  const int half = lane >> 4;
  const int koff = half * 2;

  const int cp = wave % coutPairs;
  const int pt = wave / coutPairs;

  // decompose output pixel row r -> (b, to, mo); layout [B, To, Mo, COUT]
  const int r = pt * 16 + l;
  const int ToMo = To * Mo;
  const int b = r / ToMo;
  int rem = r - b * ToMo;
  const int to = rem / Mo;
  const int mo = rem - to * Mo;

  const int ncol0 = cp * 32 + l;  // first output channel column for this lane
  const int ncol1 = ncol0 + 16;   // second tile's column

  v8f acc0 = V8ZERO, acc1 = V8ZERO;
  for (int ky = 0; ky < 3; ++ky) {
    const int ti = to * 2 + ky - padT;
    const bool vy = (ti >= 0) && (ti < Ti);
    for (int kx = 0; kx < 3; ++kx) {
      const int mi = mo * 2 + kx - padM;
      const bool vld = vy && (mi >= 0) && (mi < Mi);
      const float* aptr = in + (size_t)((b * Ti + ti) * Mi + mi) * CIN;
      const float* bp0 = w + (size_t)((ky * 3 + kx) * CIN) * COUT + ncol0;
      for (int kb = 0; kb < CIN; kb += 4) {
        const int k0 = kb + koff;
        v2f A, B0, B1;
        A.x = (vld && (k0 < CIN)) ? aptr[k0] : 0.0f;
        A.y = (vld && (k0 + 1 < CIN)) ? aptr[k0 + 1] : 0.0f;
        const float* bpk0 = bp0 + (size_t)k0 * COUT;
        const float* bpk1 = bp0 + (size_t)(k0 + 1) * COUT;
        B0.x = (k0 < CIN) ? bpk0[0] : 0.0f;
        B0.y = (k0 + 1 < CIN) ? bpk1[0] : 0.0f;
        B1.x = (k0 < CIN) ? bpk0[16] : 0.0f;
        B1.y = (k0 + 1 < CIN) ? bpk1[16] : 0.0f;
        acc0 = wmma4(A, B0, acc0);
        acc1 = wmma4(A, B1, acc1);
      }
    }
  }

  // fold conv-bias into BN affine: y = acc*sc + (bias-mean)*sc + beta
  const float sc0 = gamma[ncol0] * rsqrtf(var[ncol0] + 1e-3f);
  const float sh0 = beta[ncol0] + (bias[ncol0] - mean[ncol0]) * sc0;
  const float sc1 = gamma[ncol1] * rsqrtf(var[ncol1] + 1e-3f);
  const float sh1 = beta[ncol1] + (bias[ncol1] - mean[ncol1]) * sc1;
#pragma unroll
  for (int g = 0; g < 8; ++g) {
    const int ro = pt * 16 + g + half * 8;  // D row: lanes0-15 -> g, 16-31 -> 8+g
    float y0 = acc0[g] * sc0 + sh0;
    float y1 = acc1[g] * sc1 + sh1;
    out[(size_t)ro * COUT + ncol0] = y0 > 0.0f ? y0 : 0.0f;
    out[(size_t)ro * COUT + ncol1] = y1 > 0.0f ? y1 : 0.0f;
  }
}

// ---------------------------------------------------------------------------
// Generic fp32 WMMA GEMM: C[M,N] = act(A[M,K] @ B[K,N] + bias[N])
// M%16==0, N%32==0, K%4==0. One wave = 16x32 tile (two C tiles, shared A).
// ACT: 0=none, 1=tanh.
// ---------------------------------------------------------------------------
template <int ACT>
__global__ __launch_bounds__(256) void gemm_bias_wmma(
    const float* __restrict__ A, const float* __restrict__ B,
    const float* __restrict__ bias, float* __restrict__ C, int M, int N,
    int K) {
  const int wave = (blockIdx.x * blockDim.x + threadIdx.x) >> 5;
  const int nPairs = N >> 5;
  const int nTiles = (M >> 4) * nPairs;
  if (wave >= nTiles) return;

  const int lane = threadIdx.x & 31;
  const int l = lane & 15;
  const int half = lane >> 4;
  const int koff = half * 2;

  const int np = wave % nPairs;
  const int mt = wave / nPairs;
  const float* arow = A + (size_t)(mt * 16 + l) * K;
  const int ncol0 = np * 32 + l;
  const int ncol1 = ncol0 + 16;

  v8f acc0 = V8ZERO, acc1 = V8ZERO;
  for (int kb = 0; kb < K; kb += 4) {
    const int k0 = kb + koff;
    v2f a, b0, b1;
    a.x = arow[k0];
    a.y = arow[k0 + 1];
    const float* bk0 = B + (size_t)k0 * N + ncol0;
    const float* bk1 = B + (size_t)(k0 + 1) * N + ncol0;
    b0.x = bk0[0];
    b0.y = bk1[0];
    b1.x = bk0[16];
    b1.y = bk1[16];
    acc0 = wmma4(a, b0, acc0);
    acc1 = wmma4(a, b1, acc1);
  }
  const float bs0 = bias ? bias[ncol0] : 0.0f;
  const float bs1 = bias ? bias[ncol1] : 0.0f;
#pragma unroll
  for (int g = 0; g < 8; ++g) {
    const int ro = mt * 16 + g + half * 8;
    float y0 = acc0[g] + bs0;
    float y1 = acc1[g] + bs1;
    if (ACT == 1) {
      y0 = tanhf(y0);
      y1 = tanhf(y1);
    }
    C[(size_t)ro * N + ncol0] = y0;
    C[(size_t)ro * N + ncol1] = y1;
  }
}

// ---------------------------------------------------------------------------
// Persistent GRU scan (Keras reset_after=True), 16 steps, B=64, units=128.
// One workgroup of 1024 threads = 32 waves; wave w owns batch-tile i=w>>3 and
// unit-tile j=w&7, computing the z/r/h 16x16 blocks of MI = h @ rkernel with
// 3 WMMA accumulators per k-step (shared A from LDS); gates applied in C-tile
// lane layout. MX (= x@kernel + bi) precomputed; br added here.
// ---------------------------------------------------------------------------
__global__ __launch_bounds__(1024) void gru_scan_wmma(
    const float* __restrict__ MX,  // [64*16, 384], row = b*16 + t
    const float* __restrict__ rk,  // [128, 384]
    const float* __restrict__ br,  // [384] recurrent bias
    float* __restrict__ hT) {      // [64, 128]
  __shared__ float h[64][132];
  const int tid = threadIdx.x;
  for (int idx = tid; idx < 64 * 128; idx += 1024)
    h[idx >> 7][idx & 127] = 0.0f;
  __syncthreads();

  const int wv = tid >> 5;
  const int lane = tid & 31;
  const int l = lane & 15;
  const int half = lane >> 4;
  const int koff = half * 2;
  const int i = wv >> 3;  // batch tile 0..3
  const int j = wv & 7;   // unit tile 0..7
  const int arow = i * 16 + l;
  const int u = j * 16 + l;  // unit column 0..127
  const float brz = br[u], brr = br[128 + u], brh = br[256 + u];

  for (int t = 0; t < 16; ++t) {
    v8f cz = V8ZERO, cr = V8ZERO, ch = V8ZERO;
    for (int kb = 0; kb < 128; kb += 4) {
      const int k0 = kb + koff;
      v2f a;
      a.x = h[arow][k0];
      a.y = h[arow][k0 + 1];
      const float* r0 = rk + (size_t)k0 * 384;
      const float* r1 = rk + (size_t)(k0 + 1) * 384;
      v2f bz, brv, bh;
      bz.x = r0[u];        bz.y = r1[u];
      brv.x = r0[128 + u]; brv.y = r1[128 + u];
      bh.x = r0[256 + u];  bh.y = r1[256 + u];
      cz = wmma4(a, bz, cz);
      cr = wmma4(a, brv, cr);
      ch = wmma4(a, bh, ch);
    }
    __syncthreads();  // all waves done reading h for this step
#pragma unroll
    for (int g = 0; g < 8; ++g) {
      const int bb = i * 16 + g + half * 8;  // batch row (C-tile layout)
      const float* mx = MX + (size_t)(bb * 16 + t) * 384;
      const float z = sigmoidf_(mx[u] + cz[g] + brz);
      const float r = sigmoidf_(mx[128 + u] + cr[g] + brr);
      const float hh = tanhf(mx[256 + u] + r * (ch[g] + brh));
      const float hold = h[bb][u];
      h[bb][u] = z * hold + (1.0f - z) * hh;  // tile (i,j) exclusively owned
    }
    __syncthreads();
  }
  for (int idx = tid; idx < 64 * 128; idx += 1024)
    hT[idx] = h[idx >> 7][idx & 127];
}

// ---------------------------------------------------------------------------
// GST tokens -> K,V (tiny: 10 tokens). kbuf/vbuf: [10, 256].
// ---------------------------------------------------------------------------
__global__ __launch_bounds__(256) void gst_kv(
    const float* __restrict__ gst,  // [10, 64]
    const float* __restrict__ wk, const float* __restrict__ bk,
    const float* __restrict__ wv, const float* __restrict__ bv,
    float* __restrict__ kbuf, float* __restrict__ vbuf) {
  const int j = blockIdx.x;   // token 0..9
  const int n = threadIdx.x;  // col 0..255
  float sk = bk[n], sv = bv[n];
  for (int d = 0; d < 64; ++d) {
    const float tok = tanhf(gst[j * 64 + d]);
    sk += tok * wk[d * 256 + n];
    sv += tok * wv[d * 256 + n];
  }
  kbuf[j * 256 + n] = sk;
  vbuf[j * 256 + n] = sv;
}

// ---------------------------------------------------------------------------
// Per-(batch,head) attention over 10 tokens + assemble concat_query [64,384].
// Also writes attn probabilities to d_out (second output).
// ---------------------------------------------------------------------------
__global__ __launch_bounds__(256) void attn_small(
    const float* __restrict__ q,     // [64, 256]
    const float* __restrict__ q_in,  // [64, 128]
    const float* __restrict__ kbuf,  // [10, 256]
    const float* __restrict__ vbuf,  // [10, 256]
    float* __restrict__ attn_out,    // [64, 4, 1, 10]
    float* __restrict__ concat) {    // [64, 384]
  const int tid = threadIdx.x;
  const int bb = tid >> 2, hd = tid & 3;
  const float* qv = q + bb * 256 + hd * 64;
  float lg[10];
  float mx = -1e30f;
  for (int jj = 0; jj < 10; ++jj) {
    const float* kv = kbuf + jj * 256 + hd * 64;
    float s = 0.0f;
    for (int d = 0; d < 64; ++d) s += qv[d] * kv[d];
    s *= 0.125f;  // 1/sqrt(64)
    lg[jj] = s;
    mx = fmaxf(mx, s);
  }
  float sum = 0.0f;
  for (int jj = 0; jj < 10; ++jj) {
    lg[jj] = expf(lg[jj] - mx);
    sum += lg[jj];
  }
  const float inv = 1.0f / sum;
  for (int jj = 0; jj < 10; ++jj) {
    lg[jj] *= inv;
    attn_out[(bb * 4 + hd) * 10 + jj] = lg[jj];
  }
  for (int d = 0; d < 64; ++d) {
    float o = 0.0f;
    for (int jj = 0; jj < 10; ++jj) o += lg[jj] * vbuf[jj * 256 + hd * 64 + d];
    concat[bb * 384 + 128 + hd * 64 + d] = o;
  }
  for (int c = 0; c < 32; ++c)
    concat[bb * 384 + hd * 32 + c] = q_in[bb * 128 + hd * 32 + c];
}

// ---------------------------------------------------------------------------
extern "C" void kernel_launch(void* const* d_in, const int* in_sizes, int n_in,
                              void* d_out, int out_size, void* d_ws,
                              size_t ws_size, hipStream_t stream) {
  (void)in_sizes; (void)n_in; (void)out_size; (void)ws_size;
  // setup_inputs() order, tuples expanded:
  const float* x = (const float*)d_in[0];
  const float *cw[6], *cb[6], *gm[6], *bt[6], *mn[6], *vr[6];
  for (int i = 0; i < 6; ++i) {
    cw[i] = (const float*)d_in[1 + i];
    cb[i] = (const float*)d_in[7 + i];
    gm[i] = (const float*)d_in[13 + i];
    bt[i] = (const float*)d_in[19 + i];
    mn[i] = (const float*)d_in[25 + i];
    vr[i] = (const float*)d_in[31 + i];
  }
  const float* gru_kernel = (const float*)d_in[37];   // [256, 384]
  const float* gru_rkernel = (const float*)d_in[38];  // [128, 384]
  const float* gru_bias = (const float*)d_in[39];     // [2, 384]
  const float* proj_w = (const float*)d_in[40];       // [128, 128]
  const float* proj_b = (const float*)d_in[41];
  const float* wq = (const float*)d_in[42];  // [128, 256]
  const float* bq = (const float*)d_in[43];
  const float* wk = (const float*)d_in[44];  // [64, 256]
  const float* bk = (const float*)d_in[45];
  const float* wv = (const float*)d_in[46];  // [64, 256]
  const float* bv = (const float*)d_in[47];
  const float* wo = (const float*)d_in[48];  // [384, 256]
  const float* bo = (const float*)d_in[49];
  const float* gst = (const float*)d_in[50];  // [10, 64]
  // d_in[51] = drop_n_heads (0, identity at inference)

  // workspace layout (floats): bufA = conv1 out (max), bufB = conv2 out (max)
  float* bufA = (float*)d_ws;                 // 41,943,040 f
  float* bufB = bufA + 41943040ull;           // 10,485,760 f
  float* MX     = bufA;                       // [1024,384] (bufA free post-conv6)
  float* hT     = bufA + 393216;              // [64,128]
  float* q_in   = bufA + 401408;              // [64,128]
  float* q      = bufA + 409600;              // [64,256]
  float* kbuf   = bufA + 425984;              // [10,256]
  float* vbuf   = bufA + 428544;              // [10,256]
  float* concat = bufA + 431104;              // [64,384]
  float* enc = (float*)d_out;                 // [64,1,256] = 16384 f
  float* attn = enc + 16384;                  // [64,4,1,10] = 2560 f

  // --- conv stack (NHWC, 3x3 s2 SAME), ping-pong A<->B ---
  // waves = pixTiles * (COUT/32); 8 waves per 256-thread block
  // L1: 1024x80x1 -> 512x40x32, pixTiles = 81920, waves = 81920
  conv_bn_relu_wmma<1, 32><<<10240, 256, 0, stream>>>(
      x, bufA, cw[0], cb[0], gm[0], bt[0], mn[0], vr[0],
      1024, 80, 512, 40, 0, 0, 81920);
  // L2: 512x40x32 -> 256x20x32, pixTiles = 20480, waves = 20480
  conv_bn_relu_wmma<32, 32><<<2560, 256, 0, stream>>>(
      bufA, bufB, cw[1], cb[1], gm[1], bt[1], mn[1], vr[1],
      512, 40, 256, 20, 0, 0, 20480);
  // L3: 256x20x32 -> 128x10x64, pixTiles = 5120, waves = 10240
  conv_bn_relu_wmma<32, 64><<<1280, 256, 0, stream>>>(
      bufB, bufA, cw[2], cb[2], gm[2], bt[2], mn[2], vr[2],
      256, 20, 128, 10, 0, 0, 5120);
  // L4: 128x10x64 -> 64x5x64, pixTiles = 1280, waves = 2560
  conv_bn_relu_wmma<64, 64><<<320, 256, 0, stream>>>(
      bufA, bufB, cw[3], cb[3], gm[3], bt[3], mn[3], vr[3],
      128, 10, 64, 5, 0, 0, 1280);
  // L5: 64x5x64 -> 32x3x128 (padM=1), pixTiles = 384, waves = 1536
  conv_bn_relu_wmma<64, 128><<<192, 256, 0, stream>>>(
      bufB, bufA, cw[4], cb[4], gm[4], bt[4], mn[4], vr[4],
      64, 5, 32, 3, 0, 1, 384);
  // L6: 32x3x128 -> 16x2x128 (padM=1), pixTiles = 128, waves = 512
  conv_bn_relu_wmma<128, 128><<<64, 256, 0, stream>>>(
      bufA, bufB, cw[5], cb[5], gm[5], bt[5], mn[5], vr[5],
      32, 3, 16, 2, 0, 1, 128);

  // --- GRU input projection for all timesteps: MX = X@kernel + bi ---
  // X = conv6 out viewed as [64*16, 256]; waves = 64 * 12 = 768 -> 96 blocks
  gemm_bias_wmma<0><<<96, 256, 0, stream>>>(bufB, gru_kernel, gru_bias, MX,
                                            1024, 384, 256);
  // --- recurrent scan (persistent, 32 waves) ---
  gru_scan_wmma<<<1, 1024, 0, stream>>>(MX, gru_rkernel, gru_bias + 384, hT);

  // --- q_in = tanh(hT @ proj_w + proj_b) : 16 waves -> 2 blocks ---
  gemm_bias_wmma<1><<<2, 256, 0, stream>>>(hT, proj_w, proj_b, q_in, 64, 128,
                                           128);
  // --- q = q_in @ wq + bq : 32 waves -> 4 blocks ---
  gemm_bias_wmma<0><<<4, 256, 0, stream>>>(q_in, wq, bq, q, 64, 256, 128);
  // --- K,V from GST tokens ---
  gst_kv<<<10, 256, 0, stream>>>(gst, wk, bk, wv, bv, kbuf, vbuf);
  // --- softmax attention + concat assembly (writes attn output) ---
  attn_small<<<1, 256, 0, stream>>>(q, q_in, kbuf, vbuf, attn, concat);
  // --- enc_out = concat @ wo + bo ---
  gemm_bias_wmma<0><<<4, 256, 0, stream>>>(concat, wo, bo, enc, 64, 256, 384);
}